// GNN_GCN_86294482911408
// MI455X (gfx1250) — compile-verified
//
#include <hip/hip_runtime.h>
#include <hip/hip_bf16.h>
#include <math.h>

typedef __attribute__((ext_vector_type(2))) float v2f;
typedef __attribute__((ext_vector_type(8))) float v8f;

#define N_NODES 50000
#define N_EDGES 500000
#define N_GRAPHS 64
#define IN_DIM 64
#define HID 128
#define MID 64
#define BN_EPS 1e-5f

// ---------------------------------------------------------------------------
// Degree / normalization
// ---------------------------------------------------------------------------
__global__ void k_deg_init(float* deg, int n) {
    int i = blockIdx.x * blockDim.x + threadIdx.x;
    if (i < n) deg[i] = 1.0f;                       // self-loop contributes 1
}

__global__ void k_deg_edges(const int* __restrict__ ei, float* deg, int E) {
    int e = blockIdx.x * blockDim.x + threadIdx.x;
    if (e < E) atomicAdd(&deg[ei[E + e]], 1.0f);    // dst = ei[1][e]
}

__global__ void k_dinv(float* deg, int n) {
    int i = blockIdx.x * blockDim.x + threadIdx.x;
    if (i < n) deg[i] = rsqrtf(deg[i]);             // in place: deg -> dinv
}

// ---------------------------------------------------------------------------
// WMMA f32 GEMM: out[rows x 128] = in[rows x kdim] @ W[kdim x 128] (+ bias)
// Block = 256 threads = 8 waves; each block handles a 16-row slab; wave w
// computes the 16x16 tile at columns [16w, 16w+16). A slab staged in LDS
// with stride 132 floats (4 mod 64 banks -> conflict-free).
// EXEC is all-ones for every thread (no divergence) as WMMA requires.
// ---------------------------------------------------------------------------
__global__ void k_gemm_wmma(const float* __restrict__ in,
                            const float* __restrict__ W,
                            const float* __restrict__ bias,
                            float* __restrict__ out,
                            int kdim) {
    __shared__ float As[16][132];
    const int row0 = blockIdx.x * 16;
    const int tid  = threadIdx.x;

    // cooperative stage of the 16 x kdim A-slab
    const int total = 16 * kdim;
    for (int idx = tid; idx < total; idx += 256) {
        int r = idx / kdim;
        int c = idx - r * kdim;
        As[r][c] = in[(long long)(row0 + r) * kdim + c];
    }
    __syncthreads();

    const int lane = tid & 31;
    const int wv   = tid >> 5;        // 0..7 -> column tile
    const int n0   = wv * 16;
    const int nl   = lane & 15;
    const int hi   = lane >> 4;       // lane-half selects K pair / M half
    const int m    = lane & 15;

    __builtin_prefetch(&W[n0 + nl], 0, 3);   // global_prefetch_b8 on B panel

    v8f c = {};
    for (int k = 0; k < kdim; k += 4) {
        const int kb = k + 2 * hi;
        v2f a, b;
        a.x = As[m][kb];
        a.y = As[m][kb + 1];
        b.x = W[(long long)kb * HID + n0 + nl];
        b.y = W[(long long)(kb + 1) * HID + n0 + nl];
        // V_WMMA_F32_16X16X4_F32 : D = A(16x4) * B(4x16) + C, full f32
        c = __builtin_amdgcn_wmma_f32_16x16x4_f32(
                false, a, false, b, (short)0, c, false, false);
    }

    const float bv = bias ? bias[n0 + nl] : 0.0f;
    #pragma unroll
    for (int r = 0; r < 8; ++r) {
        const int mr = r + 8 * hi;    // C/D layout: VGPR r holds M=r (lo) / M=8+r (hi)
        out[(long long)(row0 + mr) * HID + n0 + nl] = c[r] + bv;
    }
}

// ---------------------------------------------------------------------------
// Self-loop init: agg[i,f] = bc[f] + t[i,f] * dinv[i]^2
// ---------------------------------------------------------------------------
__global__ void k_selfloop_init(const float* __restrict__ t,
                                const float* __restrict__ dinv,
                                const float* __restrict__ bc,
                                float* __restrict__ agg) {
    long long tid = (long long)blockIdx.x * blockDim.x + threadIdx.x;
    if (tid >= (long long)N_NODES * HID) return;
    int i = (int)(tid >> 7);
    int f = (int)(tid & 127);
    float di = dinv[i];
    agg[tid] = bc[f] + t[tid] * di * di;
}

// ---------------------------------------------------------------------------
// Edge scatter: one wave per edge; lane handles a float4 feature block.
// Gather of t[src] is one coalesced 512B wave transaction; scatter is
// global_atomic_add_f32 into the L2-resident agg buffer.
// ---------------------------------------------------------------------------
__global__ void k_edge_scatter(const float* __restrict__ t,
                               const int* __restrict__ ei,
                               const float* __restrict__ dinv,
                               float* __restrict__ agg,
                               int E) {
    long long tid = (long long)blockIdx.x * blockDim.x + threadIdx.x;
    int e  = (int)(tid >> 5);
    int fb = (int)((tid & 31) << 2);
    if (e >= E) return;
    int s = ei[e];
    int d = ei[E + e];
    float nrm = dinv[s] * dinv[d];
    const float4 v = *(const float4*)(t + (long long)s * HID + fb);
    float* dst = agg + (long long)d * HID + fb;
    atomicAdd(dst + 0, v.x * nrm);
    atomicAdd(dst + 1, v.y * nrm);
    atomicAdd(dst + 2, v.z * nrm);
    atomicAdd(dst + 3, v.w * nrm);
}

// ---------------------------------------------------------------------------
// Batch-norm: stats (sum, sumsq per feature) then fused apply + ReLU
// ---------------------------------------------------------------------------
__global__ void k_zero(float* p, int n) {
    int i = blockIdx.x * blockDim.x + threadIdx.x;
    if (i < n) p[i] = 0.0f;
}

__global__ void k_bn_stats(const float* __restrict__ h, float* __restrict__ stats) {
    // blockDim = 128 (thread = feature); block strides over node rows
    const int f = threadIdx.x;
    float s = 0.0f, s2 = 0.0f;
    for (int i = blockIdx.x; i < N_NODES; i += gridDim.x) {
        float v = h[(long long)i * HID + f];
        s  += v;
        s2 += v * v;
    }
    atomicAdd(&stats[f], s);
    atomicAdd(&stats[HID + f], s2);
}

__global__ void k_bn_apply_relu(float* __restrict__ h,
                                const float* __restrict__ stats,
                                const float* __restrict__ gamma,
                                const float* __restrict__ beta) {
    long long tid = (long long)blockIdx.x * blockDim.x + threadIdx.x;
    if (tid >= (long long)N_NODES * HID) return;
    int f = (int)(tid & 127);
    const float invN = 1.0f / (float)N_NODES;
    float mu  = stats[f] * invN;
    float var = stats[HID + f] * invN - mu * mu;
    float v = (h[tid] - mu) * rsqrtf(var + BN_EPS) * gamma[f] + beta[f];
    h[tid] = fmaxf(v, 0.0f);
}

// ---------------------------------------------------------------------------
// Global mean pool + MLP head
// ---------------------------------------------------------------------------
__global__ void k_pool(const float* __restrict__ h,
                       const int* __restrict__ batch,
                       float* __restrict__ pooled,
                       float* __restrict__ cnt) {
    long long tid = (long long)blockIdx.x * blockDim.x + threadIdx.x;
    if (tid >= (long long)N_NODES * HID) return;
    int i = (int)(tid >> 7);
    int f = (int)(tid & 127);
    int g = batch[i];
    atomicAdd(&pooled[(long long)g * HID + f], h[tid]);
    if (f == 0) atomicAdd(&cnt[g], 1.0f);
}

__global__ void k_mlp(const float* __restrict__ pooled,
                      const float* __restrict__ cnt,
                      const float* __restrict__ W1, const float* __restrict__ b1,
                      const float* __restrict__ W2, const float* __restrict__ b2,
                      float* __restrict__ out) {
    __shared__ float zsh[MID];
    const int g = blockIdx.x;
    const int j = threadIdx.x;           // 0..63 hidden units
    const float inv = 1.0f / fmaxf(cnt[g], 1.0f);
    float acc = b1[j];
    for (int k = 0; k < HID; ++k)
        acc += pooled[(long long)g * HID + k] * inv * W1[k * MID + j];
    zsh[j] = fmaxf(acc, 0.0f);
    __syncthreads();
    if (j == 0) {
        float s = b2[0];
        for (int k = 0; k < MID; ++k) s += zsh[k] * W2[k];
        out[g] = 1.0f / (1.0f + expf(-s));
    }
}

// ---------------------------------------------------------------------------
// Launch
// ---------------------------------------------------------------------------
extern "C" void kernel_launch(void* const* d_in, const int* in_sizes, int n_in,
                              void* d_out, int out_size, void* d_ws, size_t ws_size,
                              hipStream_t stream) {
    const float* x      = (const float*)d_in[0];
    const int*   ei     = (const int*)  d_in[1];   // [2, E] flat
    const int*   batch  = (const int*)  d_in[2];
    const float* W_enc  = (const float*)d_in[3];
    const float* b_enc  = (const float*)d_in[4];
    const float* Wc     = (const float*)d_in[5];   // [2,128,128]
    const float* bc     = (const float*)d_in[6];   // [2,128]
    const float* gamma  = (const float*)d_in[7];
    const float* beta   = (const float*)d_in[8];
    const float* W1     = (const float*)d_in[9];
    const float* b1     = (const float*)d_in[10];
    const float* W2     = (const float*)d_in[11];
    const float* b2     = (const float*)d_in[12];
    float* out = (float*)d_out;

    // workspace layout (256B aligned)
    char* base = (char*)d_ws;
    const size_t HB = (size_t)N_NODES * HID * sizeof(float);          // 25.6 MB
    float* H     = (float*)(base);
    float* T     = (float*)(base + HB);
    float* DINV  = (float*)(base + 2 * HB);                            // 200 KB
    float* STATS = (float*)(base + 2 * HB + ((N_NODES * 4 + 255) & ~255ull));
    float* POOL  = (float*)((char*)STATS + 1024);
    float* CNT   = (float*)((char*)POOL + (size_t)N_GRAPHS * HID * sizeof(float));

    const int ELEMS = N_NODES * HID;

    // 1) degree -> dinv
    k_deg_init <<<(N_NODES + 255) / 256, 256, 0, stream>>>(DINV, N_NODES);
    k_deg_edges<<<(N_EDGES + 255) / 256, 256, 0, stream>>>(ei, DINV, N_EDGES);
    k_dinv     <<<(N_NODES + 255) / 256, 256, 0, stream>>>(DINV, N_NODES);

    // 2) encoder: H = x @ W_enc + b_enc   (WMMA, K = 64)
    k_gemm_wmma<<<N_NODES / 16, 256, 0, stream>>>(x, W_enc, b_enc, H, IN_DIM);

    // 3) two GCN layers
    for (int l = 0; l < 2; ++l) {
        const float* Wl = Wc + (size_t)l * HID * HID;
        const float* bl = bc + (size_t)l * HID;
        // T = H @ Wl (WMMA, K = 128)
        k_gemm_wmma<<<N_NODES / 16, 256, 0, stream>>>(H, Wl, nullptr, T, HID);
        // H = bl + T * dinv^2  (self-loops), then scatter edges into H
        k_selfloop_init<<<(ELEMS + 255) / 256, 256, 0, stream>>>(T, DINV, bl, H);
        long long ethreads = (long long)N_EDGES * 32;
        k_edge_scatter<<<(int)((ethreads + 255) / 256), 256, 0, stream>>>(T, ei, DINV, H, N_EDGES);
        // batch norm + relu (in place on H)
        k_zero<<<1, 256, 0, stream>>>(STATS, 2 * HID);
        k_bn_stats<<<512, 128, 0, stream>>>(H, STATS);
        k_bn_apply_relu<<<(ELEMS + 255) / 256, 256, 0, stream>>>(
            H, STATS, gamma + (size_t)l * HID, beta + (size_t)l * HID);
    }

    // 4) pool + MLP head
    k_zero<<<(N_GRAPHS * HID + N_GRAPHS + 255) / 256, 256, 0, stream>>>(POOL, N_GRAPHS * HID + N_GRAPHS);
    // CNT sits directly after POOL, zeroed by the same call
    k_pool<<<(ELEMS + 255) / 256, 256, 0, stream>>>(H, batch, POOL, CNT);
    k_mlp<<<N_GRAPHS, MID, 0, stream>>>(POOL, CNT, W1, b1, W2, b2, out);
}